// MultiResidualAttentionUpsamplerEdgeDecoder_41918880808990
// MI455X (gfx1250) — compile-verified
//
#include <hip/hip_runtime.h>
#include <hip/hip_bf16.h>

typedef float v2f __attribute__((ext_vector_type(2)));
typedef float v8f __attribute__((ext_vector_type(8)));

__device__ __forceinline__ float lrelu_(float v) { return v >= 0.f ? v : 0.2f * v; }

// ---------------------------------------------------------------------------
// Implicit-GEMM conv via V_WMMA_F32_16X16X4_F32 with 2x2 register blocking.
// GEMM view: M = Cout, N = Hout*Wout, K = ks*ks*Cin with ordering
//            k = tap*Cin + ci  (tap = kh*ks+kw)  -- order-invariant sum,
//            chosen so the inner loop over ci has pure strided addressing.
// One wave (32 threads) per 32x32 output macro-tile (4 WMMA accumulators):
// per K-step of 4: 4 A loads + 4 B loads feed 4 WMMAs (1 wmma / 2 loads).
// Requires Cout%32==0, Npix%32==0, Cin%16==0 (true for all call sites).
// A (16x4, f32): lanes 0-15 rows M=0..15; VGPR0 = K(+0 / +2 for hi lanes),
//                VGPR1 = K(+1 / +3).  B (4x16) mirrored; C/D standard 8-VGPR.
// ---------------------------------------------------------------------------
__global__ __launch_bounds__(32)
void conv_wmma_f32(const float* __restrict__ in, const float* __restrict__ wgt,
                   const float* __restrict__ bias, float* __restrict__ out,
                   int Cin, int Hin, int Win, int Cout, int Hout, int Wout,
                   int ks, int stride, int pad, int do_lrelu)
{
  const int lane = threadIdx.x;
  const int half = lane >> 4;        // 0: lanes 0-15, 1: lanes 16-31
  const int l    = lane & 15;
  const int tileN = blockIdx.x << 5;
  const int tileM = blockIdx.y << 5;
  const int kk    = ks * ks;
  const int Npix  = Hout * Wout;
  const int HWin  = Hin * Win;

  // two N columns per lane (N tile 0 and N tile 1)
  const int n0 = tileN + l;
  const int n1 = n0 + 16;
  const int oh0 = n0 / Wout, ow0 = n0 - oh0 * Wout;
  const int oh1 = n1 / Wout, ow1 = n1 - oh1 * Wout;
  const int ohs0 = oh0 * stride - pad, ows0 = ow0 * stride - pad;
  const int ohs1 = oh1 * stride - pad, ows1 = ow1 * stride - pad;

  // two M rows per lane (M tile 0 and M tile 1): w[(m*Cin + ci)*kk + tap]
  const float* wbase0 = wgt + (size_t)(tileM + l) * Cin * kk;
  const float* wbase1 = wgt + (size_t)(tileM + 16 + l) * Cin * kk;

  v8f acc00 = {}, acc01 = {}, acc10 = {}, acc11 = {};

  for (int tap = 0; tap < kk; ++tap) {
    const int kh = tap / ks, kw = tap - kh * ks;     // once per tap, cheap
    const int ih0 = ohs0 + kh, iw0 = ows0 + kw;
    const int ih1 = ohs1 + kh, iw1 = ows1 + kw;
    const bool inb0 = (unsigned)ih0 < (unsigned)Hin && (unsigned)iw0 < (unsigned)Win;
    const bool inb1 = (unsigned)ih1 < (unsigned)Hin && (unsigned)iw1 < (unsigned)Win;
    // branchless: always load from a valid address, multiply by 0/1 mask
    const float  m0 = inb0 ? 1.f : 0.f;
    const float  m1 = inb1 ? 1.f : 0.f;
    const float* bp0 = inb0 ? (in + (size_t)ih0 * Win + iw0) : in;
    const float* bp1 = inb1 ? (in + (size_t)ih1 * Win + iw1) : in;
    const float* wp0 = wbase0 + tap;
    const float* wp1 = wbase1 + tap;

    for (int ci = 0; ci < Cin; ci += 16) {
#pragma unroll
      for (int u = 0; u < 4; ++u) {
        const int c0 = ci + (u << 2) + (half << 1);
        v2f a0, a1, b0, b1;
        a0.x = wp0[(size_t)c0 * kk];
        a0.y = wp0[(size_t)(c0 + 1) * kk];
        a1.x = wp1[(size_t)c0 * kk];
        a1.y = wp1[(size_t)(c0 + 1) * kk];
        b0.x = m0 * bp0[(size_t)c0 * HWin];
        b0.y = m0 * bp0[(size_t)(c0 + 1) * HWin];
        b1.x = m1 * bp1[(size_t)c0 * HWin];
        b1.y = m1 * bp1[(size_t)(c0 + 1) * HWin];
        acc00 = __builtin_amdgcn_wmma_f32_16x16x4_f32(false, a0, false, b0,
                                                      (short)0, acc00, false, false);
        acc01 = __builtin_amdgcn_wmma_f32_16x16x4_f32(false, a0, false, b1,
                                                      (short)0, acc01, false, false);
        acc10 = __builtin_amdgcn_wmma_f32_16x16x4_f32(false, a1, false, b0,
                                                      (short)0, acc10, false, false);
        acc11 = __builtin_amdgcn_wmma_f32_16x16x4_f32(false, a1, false, b1,
                                                      (short)0, acc11, false, false);
      }
    }
  }

  const int ncol0 = tileN + l;          // C/D: col = lane&15, row = r + 8*half
#pragma unroll
  for (int r = 0; r < 8; ++r) {
    const int mo0 = tileM + r + (half << 3);
    const int mo1 = mo0 + 16;
    float v00 = acc00[r] + bias[mo0];
    float v01 = acc01[r] + bias[mo0];
    float v10 = acc10[r] + bias[mo1];
    float v11 = acc11[r] + bias[mo1];
    if (do_lrelu) {
      v00 = lrelu_(v00); v01 = lrelu_(v01);
      v10 = lrelu_(v10); v11 = lrelu_(v11);
    }
    out[(size_t)mo0 * Npix + ncol0]      = v00;
    out[(size_t)mo0 * Npix + ncol0 + 16] = v01;
    out[(size_t)mo1 * Npix + ncol0]      = v10;
    out[(size_t)mo1 * Npix + ncol0 + 16] = v11;
  }
}

// ---------------------------------------------------------------------------
// Direct convolution (tiny-K cases: Cin==1 multi-res convs, enc0, cls head)
// ---------------------------------------------------------------------------
__global__ void conv_direct(const float* __restrict__ in, const float* __restrict__ wgt,
                            const float* __restrict__ bias, float* __restrict__ out,
                            int Cin, int H, int W, int Cout, int ks, int pad, int do_lrelu)
{
  int idx = blockIdx.x * blockDim.x + threadIdx.x;
  int total = Cout * H * W;
  if (idx >= total) return;
  int n  = idx % (H * W);
  int co = idx / (H * W);
  int oh = n / W, ow = n - oh * W;
  float s = bias[co];
  for (int ci = 0; ci < Cin; ++ci) {
    const float* ip = in + (size_t)ci * H * W;
    const float* wp = wgt + ((size_t)co * Cin + ci) * ks * ks;
    for (int kh = 0; kh < ks; ++kh) {
      int ih = oh - pad + kh;
      if ((unsigned)ih >= (unsigned)H) continue;
      for (int kw = 0; kw < ks; ++kw) {
        int iw = ow - pad + kw;
        if ((unsigned)iw >= (unsigned)W) continue;
        s += wp[kh * ks + kw] * ip[ih * W + iw];
      }
    }
  }
  if (do_lrelu) s = lrelu_(s);
  out[idx] = s;
}

// ---------------------------------------------------------------------------
// BatchNorm (training mode): per-channel biased mean/var, then normalize.
// ---------------------------------------------------------------------------
__global__ __launch_bounds__(256)
void bn_stats(const float* __restrict__ x, float* __restrict__ mean,
              float* __restrict__ var, int Npix)
{
  __shared__ float ssum[256];
  __shared__ float ssq[256];
  int c = blockIdx.x;
  int tid = threadIdx.x;
  const float* xp = x + (size_t)c * Npix;
  float s = 0.f, q = 0.f;
  for (int i = tid; i < Npix; i += 256) {
    float v = xp[i];
    s += v;
    q += v * v;
  }
  ssum[tid] = s; ssq[tid] = q;
  __syncthreads();
  for (int st = 128; st > 0; st >>= 1) {
    if (tid < st) { ssum[tid] += ssum[tid + st]; ssq[tid] += ssq[tid + st]; }
    __syncthreads();
  }
  if (tid == 0) {
    float m = ssum[0] / (float)Npix;
    mean[c] = m;
    var[c]  = ssq[0] / (float)Npix - m * m;
  }
}

__global__ void bn_apply(const float* __restrict__ x, float* __restrict__ y,
                         const float* __restrict__ mean, const float* __restrict__ var,
                         const float* __restrict__ g, const float* __restrict__ b,
                         int Npix, int total)
{
  int idx = blockIdx.x * blockDim.x + threadIdx.x;
  if (idx >= total) return;
  int c = idx / Npix;
  float inv = rsqrtf(var[c] + 1e-5f);
  y[idx] = g[c] * (x[idx] - mean[c]) * inv + b[c];
}

// ---------------------------------------------------------------------------
// z = ctx + up2(v): nearest 2x upsample of low-res conv output + skip add
// ---------------------------------------------------------------------------
__global__ void up2_add(const float* __restrict__ ctx, const float* __restrict__ v,
                        float* __restrict__ z, int C, int H, int W)
{
  int idx = blockIdx.x * blockDim.x + threadIdx.x;
  int total = C * H * W;
  if (idx >= total) return;
  int n = idx % (H * W);
  int c = idx / (H * W);
  int oh = n / W, ow = n - oh * W;
  int hl = H >> 1, wl = W >> 1;
  z[idx] = ctx[idx] + v[((size_t)c * hl + (oh >> 1)) * wl + (ow >> 1)];
}

// ---------------------------------------------------------------------------
// Decoder tanh-additive attention, one context at a time.
// One wave per pixel; lane = channel (C == 32 == warpSize).
// scores_k = agg_b + sum_c aggw[c] * tanh(keys_pad[c,k] + q[c]);
// softmax over 9 neighbors; attended[c] = sum_k w_k * ctx_pad[c,k].
// Zero-padded neighborhood (matches nn.Unfold(3, padding=1)).
// ---------------------------------------------------------------------------
__global__ __launch_bounds__(256)
void dec_attn(const float* __restrict__ ctx, const float* __restrict__ keys,
              const float* __restrict__ q, const float* __restrict__ aggw,
              const float* __restrict__ aggb, float* __restrict__ out,
              int H, int W)
{
  int wid  = threadIdx.x >> 5;
  int lane = threadIdx.x & 31;
  int p = blockIdx.x * 8 + wid;
  int HW = H * W;
  int h = p / W, w = p - h * W;

  float qv = q[(size_t)lane * HW + p];
  float aw = aggw[lane];
  float ab = aggb[0];

  float scores[9];
#pragma unroll
  for (int k = 0; k < 9; ++k) {
    int dh = k / 3 - 1, dw = k % 3 - 1;
    int hh = h + dh, ww = w + dw;
    float kv = 0.f;
    if ((unsigned)hh < (unsigned)H && (unsigned)ww < (unsigned)W)
      kv = keys[(size_t)lane * HW + hh * W + ww];
    float t = tanhf(kv + qv) * aw;
#pragma unroll
    for (int off = 16; off > 0; off >>= 1) t += __shfl_xor(t, off, 32);
    scores[k] = t + ab;
  }

  float mx = scores[0];
#pragma unroll
  for (int k = 1; k < 9; ++k) mx = fmaxf(mx, scores[k]);
  float wts[9];
  float sum = 0.f;
#pragma unroll
  for (int k = 0; k < 9; ++k) { wts[k] = __expf(scores[k] - mx); sum += wts[k]; }
  float inv = 1.f / sum;

  float acc = 0.f;
#pragma unroll
  for (int k = 0; k < 9; ++k) {
    int dh = k / 3 - 1, dw = k % 3 - 1;
    int hh = h + dh, ww = w + dw;
    float cv = 0.f;
    if ((unsigned)hh < (unsigned)H && (unsigned)ww < (unsigned)W)
      cv = ctx[(size_t)lane * HW + hh * W + ww];
    acc += wts[k] * inv * cv;
  }
  out[(size_t)lane * HW + p] = acc;
}

// ---------------------------------------------------------------------------
// Host-side orchestration
// ---------------------------------------------------------------------------
static inline void launch_conv_wmma(hipStream_t s, const float* in, const float* w,
                                    const float* b, float* out, int Cin, int Hin, int Win,
                                    int Cout, int Hout, int Wout, int ks, int stride,
                                    int pad, int lrelu)
{
  dim3 grid((Hout * Wout) / 32, Cout / 32);
  conv_wmma_f32<<<grid, dim3(32), 0, s>>>(in, w, b, out, Cin, Hin, Win, Cout,
                                          Hout, Wout, ks, stride, pad, lrelu);
}

static inline void launch_bn(hipStream_t s, const float* x, float* y, int C, int Npix,
                             const float* g, const float* b, float* m, float* v)
{
  bn_stats<<<dim3(C), dim3(256), 0, s>>>(x, m, v, Npix);
  int tot = C * Npix;
  bn_apply<<<dim3((tot + 255) / 256), dim3(256), 0, s>>>(x, y, m, v, g, b, Npix, tot);
}

extern "C" void kernel_launch(void* const* d_in, const int* in_sizes, int n_in,
                              void* d_out, int out_size, void* d_ws, size_t ws_size,
                              hipStream_t stream)
{
  (void)in_sizes; (void)n_in; (void)out_size;
  const float* X    = (const float*)d_in[0];   // (1,1,256,256)
  const float* EDGE = (const float*)d_in[1];   // (1,1,256,256)
#define P(i) ((const float*)d_in[i])
  // params in setup_inputs insertion order, starting at index 2
  const float *enc0_w = P(2),  *enc0_b = P(3);
  const float *enc1_w = P(4),  *enc1_b = P(5);
  const float *enc2_w = P(6),  *enc2_b = P(7);
  const float *enc3_w = P(8),  *enc3_b = P(9);
  const float *enc4_w = P(10), *enc4_b = P(11);
  const float *enc5_w = P(12), *enc5_b = P(13);
  const float *mr3_w  = P(14), *mr3_b  = P(15);
  const float *mr5_w  = P(16), *mr5_b  = P(17);
  const float *mr7_w  = P(18), *mr7_b  = P(19);
  // au*_Wc (20/21,26/27,32/33,38/39,44/45) and au*_Wf (22/23,28/29,34/35,40/41,46/47)
  // are dead: softmax over a single context is identically 1.
  const float *au1_v_w = P(24), *au1_v_b = P(25);
  const float *au2_v_w = P(30), *au2_v_b = P(31);
  const float *au3_v_w = P(36), *au3_v_b = P(37);
  const float *au4_v_w = P(42), *au4_v_b = P(43);
  const float *au5_v_w = P(48), *au5_v_b = P(49);
  const float *ed_Wenc_w = P(50), *ed_Wenc_b = P(51);
  const float *ed_Wdec_w = P(52), *ed_Wdec_b = P(53);
  const float *ed_agg_w  = P(54), *ed_agg_b  = P(55);
  const float *ed_attn_w = P(56), *ed_attn_b = P(57);
  const float *ed_v_w    = P(58), *ed_v_b    = P(59);
  const float *cls_w = P(60), *cls_b = P(61);
  const float *bn2_g  = P(62), *bn2_b  = P(63);
  const float *bn32_g = P(64), *bn32_b = P(65);
  const float *bn64_g = P(66), *bn64_b = P(67);
  const float *bn128_g = P(68), *bn128_b = P(69);
  const float *bn256_g = P(70), *bn256_b = P(71);
  const float *bn512_g = P(72), *bn512_b = P(73);
  const float *mrbn_g = P(74), *mrbn_b = P(75);
#undef P

  const int HW0 = 256 * 256, HW1 = 128 * 128, HW2 = 64 * 64, HW3 = 32 * 32,
            HW4 = 16 * 16, HW5 = 8 * 8;

  float* ws = (float*)d_ws;
  size_t off = 0;
  auto alloc = [&](size_t nf) { float* p = ws + off; off += nf; return p; };

  float* e0   = alloc((size_t)32 * HW0);
  float* e1   = alloc((size_t)32 * HW0);
  float* e2   = alloc((size_t)32 * HW0);
  float* prelim = alloc((size_t)32 * HW0);     // reused as q after z5
  float* x1   = alloc((size_t)64 * HW1);       // region reused for d3/cls at the end
  float* x2   = alloc((size_t)128 * HW2);
  float* x3   = alloc((size_t)256 * HW3);
  float* x4   = alloc((size_t)512 * HW4);
  float* enc  = alloc((size_t)512 * HW5);
  float* v1   = alloc((size_t)512 * HW5);
  float* z1   = alloc((size_t)512 * HW4);
  float* v2   = alloc((size_t)256 * HW4);
  float* z2   = alloc((size_t)256 * HW3);
  float* v3   = alloc((size_t)128 * HW3);
  float* z3   = alloc((size_t)128 * HW2);
  float* v4   = alloc((size_t)64 * HW2);
  float* z4   = alloc((size_t)64 * HW1);
  float* v5   = alloc((size_t)32 * HW1);
  float* z5   = alloc((size_t)32 * HW0);       // reused as keys buffer per context
  float* cat  = alloc((size_t)128 * HW0);      // [vals(32) | attended ctx0..2 (96)]
  float* statm = alloc(512);
  float* statv = alloc(512);
  if (off * sizeof(float) > ws_size) return;   // workspace too small; bail

  float* qbuf    = prelim;   // prelim dead after z5
  float* keysbuf = z5;       // z5 dead after q/vals convs
  float* d3      = x1;       // encoder region dead after z4
  float* clsbuf  = x1 + (size_t)32 * HW0;

  // ---- multi-resolution edge branch: conv -> lrelu -> BN (mr_bn) ----------
  {
    int tot = 32 * HW0, blk = (tot + 255) / 256;
    conv_direct<<<blk, 256, 0, stream>>>(EDGE, mr3_w, mr3_b, e0, 1, 256, 256, 32, 3, 1, 1);
    launch_bn(stream, e0, e0, 32, HW0, mrbn_g, mrbn_b, statm, statv);
    conv_direct<<<blk, 256, 0, stream>>>(EDGE, mr5_w, mr5_b, e1, 1, 256, 256, 32, 5, 2, 1);
    launch_bn(stream, e1, e1, 32, HW0, mrbn_g, mrbn_b, statm, statv);
    conv_direct<<<blk, 256, 0, stream>>>(EDGE, mr7_w, mr7_b, e2, 1, 256, 256, 32, 7, 3, 1);
    launch_bn(stream, e2, e2, 32, HW0, mrbn_g, mrbn_b, statm, statv);
    // prelim: enc0 conv on x
    conv_direct<<<blk, 256, 0, stream>>>(X, enc0_w, enc0_b, prelim, 1, 256, 256, 32, 3, 1, 1);
    launch_bn(stream, prelim, prelim, 32, HW0, bn32_g, bn32_b, statm, statv);
  }

  // ---- encoder (WMMA implicit-GEMM, stride-2 3x3 convs) -------------------
  launch_conv_wmma(stream, prelim, enc1_w, enc1_b, x1, 32, 256, 256, 64, 128, 128, 3, 2, 1, 1);
  launch_bn(stream, x1, x1, 64, HW1, bn64_g, bn64_b, statm, statv);
  launch_conv_wmma(stream, x1, enc2_w, enc2_b, x2, 64, 128, 128, 128, 64, 64, 3, 2, 1, 1);
  launch_bn(stream, x2, x2, 128, HW2, bn128_g, bn128_b, statm, statv);
  launch_conv_wmma(stream, x2, enc3_w, enc3_b, x3, 128, 64, 64, 256, 32, 32, 3, 2, 1, 1);
  launch_bn(stream, x3, x3, 256, HW3, bn256_g, bn256_b, statm, statv);
  launch_conv_wmma(stream, x3, enc4_w, enc4_b, x4, 256, 32, 32, 512, 16, 16, 3, 2, 1, 1);
  launch_bn(stream, x4, x4, 512, HW4, bn512_g, bn512_b, statm, statv);
  launch_conv_wmma(stream, x4, enc5_w, enc5_b, enc, 512, 16, 16, 512, 8, 8, 3, 2, 1, 1);
  launch_bn(stream, enc, enc, 512, HW5, bn512_g, bn512_b, statm, statv);

  // ---- upsampler stages: z = ctx + up2(conv3x3(ms, v)) --------------------
  launch_conv_wmma(stream, enc, au1_v_w, au1_v_b, v1, 512, 8, 8, 512, 8, 8, 3, 1, 1, 0);
  { int t = 512 * HW4; up2_add<<<(t + 255) / 256, 256, 0, stream>>>(x4, v1, z1, 512, 16, 16); }
  launch_conv_wmma(stream, z1, au2_v_w, au2_v_b, v2, 512, 16, 16, 256, 16, 16, 3, 1, 1, 0);
  { int t = 256 * HW3; up2_add<<<(t + 255) / 256, 256, 0, stream>>>(x3, v2, z2, 256, 32, 32); }
  launch_conv_wmma(stream, z2, au3_v_w, au3_v_b, v3, 256, 32, 32, 128, 32, 32, 3, 1, 1, 0);
  { int t = 128 * HW2; up2_add<<<(t + 255) / 256, 256, 0, stream>>>(x2, v3, z3, 128, 64, 64); }
  launch_conv_wmma(stream, z3, au4_v_w, au4_v_b, v4, 128, 64, 64, 64, 64, 64, 3, 1, 1, 0);
  { int t = 64 * HW1; up2_add<<<(t + 255) / 256, 256, 0, stream>>>(x1, v4, z4, 64, 128, 128); }
  launch_conv_wmma(stream, z4, au5_v_w, au5_v_b, v5, 64, 128, 128, 32, 128, 128, 3, 1, 1, 0);
  { int t = 32 * HW0; up2_add<<<(t + 255) / 256, 256, 0, stream>>>(prelim, v5, z5, 32, 256, 256); }

  // ---- decoder attention --------------------------------------------------
  // q = 1x1 conv(z5, Wdec); vals = 3x3 conv(z5, ed_v) -> cat channels 0..31
  launch_conv_wmma(stream, z5, ed_Wdec_w, ed_Wdec_b, qbuf, 32, 256, 256, 32, 256, 256, 1, 1, 0, 0);
  launch_conv_wmma(stream, z5, ed_v_w, ed_v_b, cat, 32, 256, 256, 32, 256, 256, 3, 1, 1, 0);
  const float* ectx[3] = { e0, e1, e2 };
  for (int n = 0; n < 3; ++n) {
    launch_conv_wmma(stream, ectx[n], ed_Wenc_w, ed_Wenc_b, keysbuf,
                     32, 256, 256, 32, 256, 256, 1, 1, 0, 0);
    dec_attn<<<dim3(HW0 / 8), dim3(256), 0, stream>>>(
        ectx[n], keysbuf, qbuf, ed_agg_w, ed_agg_b,
        cat + (size_t)(32 + 32 * n) * HW0, 256, 256);
  }
  // d3 = lrelu(1x1 conv(cat, ed_attn))  (Cin=128 -> Cout=32)
  launch_conv_wmma(stream, cat, ed_attn_w, ed_attn_b, d3, 128, 256, 256, 32, 256, 256, 1, 1, 0, 1);

  // ---- classifier: 3x3 conv 32->2, lrelu, BN(2) -> d_out ------------------
  {
    int tot = 2 * HW0;
    conv_direct<<<(tot + 255) / 256, 256, 0, stream>>>(d3, cls_w, cls_b, clsbuf,
                                                       32, 256, 256, 2, 3, 1, 1);
    bn_stats<<<dim3(2), dim3(256), 0, stream>>>(clsbuf, statm, statv, HW0);
    bn_apply<<<(tot + 255) / 256, 256, 0, stream>>>(clsbuf, (float*)d_out,
                                                    statm, statv, bn2_g, bn2_b, HW0, tot);
  }
}